// PCAFreqModel_69965017252456
// MI455X (gfx1250) — compile-verified
//
#include <hip/hip_runtime.h>
#include <math.h>

// ---------------------------------------------------------------------------
// PCAFreqModel for MI455X (gfx1250): everything as fp32 WMMA 16x16x4 matmuls.
// Memory-bound (~1.3 GB min traffic -> ~57us @ 23.3 TB/s), so fp32 WMMA keeps
// full precision at zero bandwidth cost. GEMM1 uses double-buffered
// global_load_async_to_lds_b64 (ASYNCcnt) for A-tile staging; GEMM2 coalesces
// its 636 MB of output stores through LDS.
// ---------------------------------------------------------------------------

typedef __attribute__((ext_vector_type(2))) float v2f;
typedef __attribute__((ext_vector_type(8))) float v8f;

#define WMMA_F32(a, b, c) \
  __builtin_amdgcn_wmma_f32_16x16x4_f32(false, (a), false, (b), (short)0, (c), false, false)

#define NB   256      // batch
#define TSEQ 720      // seq len
#define EIN  862      // enc_in
#define RNK  32       // pca rank
#define FCUT 48       // cut freq
#define GOUT 96       // freq_out
#define NFULL 1440    // irfft length
#define MROWS (NB * TSEQ)   // 184320

// ---------------- workspace offsets (in floats) ----------------------------
#define OFF_Y     0L                                   // 5898240 (aliased by Z)
#define OFF_MEAN  (OFF_Y    + (long)MROWS * RNK)       // 8192
#define OFF_STD   (OFF_MEAN + 8192L)
#define OFF_ISTD  (OFF_STD  + 8192L)
#define OFF_XR    (OFF_ISTD + 8192L)                   // 393216
#define OFF_XI    (OFF_XR   + (long)NB * FCUT * RNK)
#define OFF_OR    (OFF_XI   + (long)NB * FCUT * RNK)   // 786432
#define OFF_OI    (OFF_OR   + (long)NB * GOUT * RNK)
#define OFF_FCOS  (OFF_OI   + (long)NB * GOUT * RNK)   // 34560
#define OFF_FSIN  (OFF_FCOS + (long)FCUT * TSEQ)
#define OFF_CCAT  (OFF_FSIN + (long)FCUT * TSEQ)       // 138240 (720 x 192)

#define TWO_PI 6.28318530717958647692f

// ---------------------------------------------------------------------------
// K0: DFT bases. Fcos[f,t]=cos(2pi f t/720), Fsin[f,t]=-sin(...)  (rfft)
//     Ccat[t, 0..95]  =  s_g cos(2pi g (t+720)/1440)
//     Ccat[t,96..191] = -s_g sin(2pi g (t+720)/1440)
//     s_0 = ratio/1440, s_g = 2*ratio/1440  (ratio=2; c2r ignores Im(X0))
// ---------------------------------------------------------------------------
__global__ __launch_bounds__(256) void k_basis(float* __restrict__ Fcos,
                                               float* __restrict__ Fsin,
                                               float* __restrict__ Ccat) {
  int idx = blockIdx.x * 256 + threadIdx.x;
  if (idx < FCUT * TSEQ) {
    int f = idx / TSEQ, t = idx % TSEQ;
    int ph = (int)(((long)f * t) % TSEQ);
    float th = TWO_PI * (float)ph / (float)TSEQ;
    Fcos[idx] = cosf(th);
    Fsin[idx] = -sinf(th);
  }
  if (idx < TSEQ * 192) {
    int t = idx / 192, c = idx % 192;
    int g = (c < GOUT) ? c : c - GOUT;
    int ph = (int)(((long)g * (t + TSEQ)) % NFULL);
    float th = TWO_PI * (float)ph / (float)NFULL;
    float s = ((g == 0) ? 2.0f : 4.0f) / (float)NFULL;  // ratio=2 folded in
    Ccat[idx] = (c < GOUT) ? (s * cosf(th)) : (-s * sinf(th));
  }
}

// ---------------------------------------------------------------------------
// Async global->LDS b64 (x_enc rows only guarantee 8B alignment: 862 % 4 = 2)
// ---------------------------------------------------------------------------
__device__ __forceinline__ void async_ld_b64(float* lp, const float* gp) {
  uint32_t loff = (uint32_t)(uintptr_t)lp;           // low 32 bits = LDS offset
  uint64_t ga   = (uint64_t)(uintptr_t)gp;
  asm volatile("global_load_async_to_lds_b64 %0, %1, off"
               :: "v"(loff), "v"(ga) : "memory");
}

// ---------------------------------------------------------------------------
// K1: Y = x_enc @ pca_encode   (184320 x 862) x (862 x 32)
// Block: 256 thr = 8 waves -> 4 M-tiles x 2 N-tiles (64 x 32 output tile).
// A staged in LDS via double-buffered async b64 loads (ASYNCcnt pipelining),
// 26 unguarded K-chunks of 32 + one guarded 30-wide epilogue chunk.
// ---------------------------------------------------------------------------
__global__ __launch_bounds__(256) void k_gemm1(const float* __restrict__ X,
                                               const float* __restrict__ P,
                                               float* __restrict__ Y) {
  __shared__ __align__(16) float sA[2][64 * 32];
  int tid = threadIdx.x;
  int wave = tid >> 5, lane = tid & 31;
  int half = lane >> 4, l16 = lane & 15;
  int mt = wave >> 1;   // 0..3
  int nt = wave & 1;    // 0..1
  long m0 = (long)blockIdx.x * 64;
  int n = nt * 16 + l16;
  v8f acc = {};

  // issue chunk 0
#pragma unroll
  for (int p = 0; p < 4; ++p) {
    int idx = p * 512 + tid * 2;
    int row = idx >> 5, col = idx & 31;
    async_ld_b64(&sA[0][idx], X + (m0 + row) * EIN + col);
  }

  for (int c = 0; c < 26; ++c) {
    int cur = c & 1;
    if (c + 1 < 26) {
#pragma unroll
      for (int p = 0; p < 4; ++p) {
        int idx = p * 512 + tid * 2;
        int row = idx >> 5, col = idx & 31;
        async_ld_b64(&sA[cur ^ 1][idx], X + (m0 + row) * EIN + (c + 1) * 32 + col);
      }
      asm volatile("s_wait_asynccnt 0x4" ::: "memory");  // chunk c landed
    } else {
      asm volatile("s_wait_asynccnt 0x0" ::: "memory");
    }
    __syncthreads();
#pragma unroll
    for (int ks = 0; ks < 32; ks += 4) {
      int kA = ks + 2 * half;
      v2f a = *(const v2f*)&sA[cur][(mt * 16 + l16) * 32 + kA];
      int kg = c * 32 + kA;
      v2f b;
      b.x = P[(long)kg * RNK + n];
      b.y = P[(long)(kg + 1) * RNK + n];
      acc = WMMA_F32(a, b, acc);
    }
    __syncthreads();  // buffer reusable
  }

  // epilogue: k = 832..861 (30 valid cols, zero-padded to 32)
#pragma unroll
  for (int i = 0; i < 8; ++i) {
    int idx = tid + i * 256;
    int row = idx >> 5, col = idx & 31;
    int k = 832 + col;
    sA[0][idx] = (k < EIN) ? X[(m0 + row) * EIN + k] : 0.0f;
  }
  __syncthreads();
#pragma unroll
  for (int ks = 0; ks < 32; ks += 4) {
    int kA = ks + 2 * half;
    v2f a = *(const v2f*)&sA[0][(mt * 16 + l16) * 32 + kA];
    int kg = 832 + kA;
    v2f b;
    b.x = (kg < EIN) ? P[(long)kg * RNK + n] : 0.0f;
    b.y = (kg + 1 < EIN) ? P[(long)(kg + 1) * RNK + n] : 0.0f;
    acc = WMMA_F32(a, b, acc);
  }

  long mrow = m0 + mt * 16;
#pragma unroll
  for (int v = 0; v < 8; ++v)
    Y[(mrow + v + 8 * half) * RNK + n] = acc[v];
}

// ---------------------------------------------------------------------------
// K2: per-(b,r) mean / std (ddof=1, +1e-5 under sqrt)
// ---------------------------------------------------------------------------
__global__ __launch_bounds__(256) void k_stats(const float* __restrict__ Y,
                                               float* __restrict__ meanv,
                                               float* __restrict__ stdv,
                                               float* __restrict__ istdv) {
  __shared__ float ss[8][32];
  __shared__ float sq[8][32];
  int b = blockIdx.x;
  int r = threadIdx.x & 31, c = threadIdx.x >> 5;
  float s = 0.0f, q = 0.0f;
  for (int j = 0; j < 90; ++j) {
    float v = Y[((long)b * TSEQ + c * 90 + j) * RNK + r];
    s += v; q += v * v;
  }
  ss[c][r] = s; sq[c][r] = q;
  __syncthreads();
  if (c == 0) {
    float S = 0.0f, Q = 0.0f;
#pragma unroll
    for (int i = 0; i < 8; ++i) { S += ss[i][r]; Q += sq[i][r]; }
    float m = S / (float)TSEQ;
    float var = (Q - S * m) / (float)(TSEQ - 1);
    float sd = sqrtf(var + 1e-5f);
    meanv[b * RNK + r] = m;
    stdv[b * RNK + r]  = sd;
    istdv[b * RNK + r] = 1.0f / sd;
  }
}

// ---------------------------------------------------------------------------
// K3: forward DFT. Per batch: Xr/Xi(48x32) = Basis(48x720) x Yn(720x32).
// 6 waves: (3 f-tiles) x (2 r-tiles); cos/sin accumulators share B fragment.
// ---------------------------------------------------------------------------
__global__ __launch_bounds__(192) void k_dft(const float* __restrict__ Y,
                                             const float* __restrict__ meanv,
                                             const float* __restrict__ istdv,
                                             const float* __restrict__ Fcos,
                                             const float* __restrict__ Fsin,
                                             float* __restrict__ Xr,
                                             float* __restrict__ Xi) {
  int b = blockIdx.x;
  int tid = threadIdx.x;
  int wave = tid >> 5, lane = tid & 31, half = lane >> 4, l16 = lane & 15;
  int f0 = (wave >> 1) * 16;
  int r  = (wave & 1) * 16 + l16;
  float mv = meanv[b * RNK + r];
  float iv = istdv[b * RNK + r];
  const float* Yb = Y + (long)b * TSEQ * RNK;
  v8f cr = {}, ci = {};
  for (int k = 0; k < TSEQ; k += 4) {
    int kk = k + 2 * half;
    v2f ac = *(const v2f*)&Fcos[(long)(f0 + l16) * TSEQ + kk];
    v2f as = *(const v2f*)&Fsin[(long)(f0 + l16) * TSEQ + kk];
    v2f bv;
    bv.x = (Yb[(long)kk * RNK + r] - mv) * iv;
    bv.y = (Yb[(long)(kk + 1) * RNK + r] - mv) * iv;
    cr = WMMA_F32(ac, bv, cr);
    ci = WMMA_F32(as, bv, ci);
  }
#pragma unroll
  for (int v = 0; v < 8; ++v) {
    int f = f0 + v + 8 * half;
    Xr[((long)b * FCUT + f) * RNK + r] = cr[v];
    Xi[((long)b * FCUT + f) * RNK + r] = ci[v];
  }
}

// ---------------------------------------------------------------------------
// K4: complex mixing  Or+iOi [b,g,r] = sum_f (Xr+iXi)[b,f,r]*(Wr+iWi)[r,g,f] + bias
// ---------------------------------------------------------------------------
__global__ __launch_bounds__(256) void k_mix(const float* __restrict__ Xr,
                                             const float* __restrict__ Xi,
                                             const float* __restrict__ Wr,
                                             const float* __restrict__ Wi,
                                             const float* __restrict__ br,
                                             const float* __restrict__ bi,
                                             float* __restrict__ Or,
                                             float* __restrict__ Oi) {
  int blk = blockIdx.x;           // b*12 + gchunk
  int b = blk / 12, gc = blk % 12;
  int r = threadIdx.x & 31, go = threadIdx.x >> 5;
  int g = gc * 8 + go;
  const float* xr = Xr + (long)b * FCUT * RNK + r;
  const float* xi = Xi + (long)b * FCUT * RNK + r;
  const float* wr = Wr + ((long)r * GOUT + g) * FCUT;
  const float* wi = Wi + ((long)r * GOUT + g) * FCUT;
  float ar = 0.0f, ai = 0.0f;
  for (int f = 0; f < FCUT; ++f) {
    float xrv = xr[(long)f * RNK], xiv = xi[(long)f * RNK];
    float wrv = wr[f], wiv = wi[f];
    ar += xrv * wrv - xiv * wiv;
    ai += xrv * wiv + xiv * wrv;
  }
  Or[((long)b * GOUT + g) * RNK + r] = ar + br[r * GOUT + g];
  Oi[((long)b * GOUT + g) * RNK + r] = ai + bi[r * GOUT + g];
}

// ---------------------------------------------------------------------------
// K5: inverse DFT + de-normalize. Per batch: Z(720x32) = Ccat(720x192) x [Or;Oi]
// then Z = Z*std + mean. 8 waves loop over 45x2 = 90 tiles. Z aliases Y.
// ---------------------------------------------------------------------------
__global__ __launch_bounds__(256) void k_idft(const float* __restrict__ Or,
                                              const float* __restrict__ Oi,
                                              const float* __restrict__ Ccat,
                                              const float* __restrict__ meanv,
                                              const float* __restrict__ stdv,
                                              float* __restrict__ Z) {
  int b = blockIdx.x;
  int tid = threadIdx.x, wave = tid >> 5, lane = tid & 31;
  int half = lane >> 4, l16 = lane & 15;
  const float* Obr = Or + (long)b * GOUT * RNK;
  const float* Obi = Oi + (long)b * GOUT * RNK;
  for (int tile = wave; tile < 90; tile += 8) {
    int t0 = (tile / 2) * 16;
    int r  = (tile % 2) * 16 + l16;
    float mv = meanv[b * RNK + r];
    float sv = stdv[b * RNK + r];
    v8f acc = {};
    for (int k = 0; k < 192; k += 4) {
      v2f a = *(const v2f*)&Ccat[(long)(t0 + l16) * 192 + k + 2 * half];
      const float* base = (k < GOUT) ? (Obr + (long)k * RNK)
                                     : (Obi + (long)(k - GOUT) * RNK);
      v2f bv;
      bv.x = base[(2 * half) * RNK + r];
      bv.y = base[(2 * half + 1) * RNK + r];
      acc = WMMA_F32(a, bv, acc);
    }
#pragma unroll
    for (int v = 0; v < 8; ++v) {
      int t = t0 + v + 8 * half;
      Z[((long)b * TSEQ + t) * RNK + r] = acc[v] * sv + mv;
    }
  }
}

// ---------------------------------------------------------------------------
// K6: out = Z @ pca_decode  (184320 x 32) x (32 x 862).
// Block: M-tile 16, 8 waves cover N=128. A (16x32) staged in LDS; C tiles go
// through LDS so the 636 MB of output is written as coalesced b64 stores
// (b128 impossible: 862 % 4 == 2 breaks 16B row alignment). Edge block
// (blockIdx.x==6) takes a block-uniform guarded path; WMMA always runs with
// EXEC all-ones (edge waves just compute zero tiles from guarded loads).
// ---------------------------------------------------------------------------
__global__ __launch_bounds__(256) void k_gemm2(const float* __restrict__ Z,
                                               const float* __restrict__ D,
                                               float* __restrict__ out) {
  __shared__ __align__(16) float sA[16 * 32];
  __shared__ __align__(16) float sC[16 * 128];
  int tid = threadIdx.x, wave = tid >> 5, lane = tid & 31;
  int half = lane >> 4, l16 = lane & 15;
  long m0 = (long)blockIdx.y * 16;
  int n0 = blockIdx.x * 128 + wave * 16;
  int n = n0 + l16;

  *(v2f*)&sA[tid * 2] = *(const v2f*)&Z[m0 * RNK + tid * 2];
  __syncthreads();

  v8f acc = {};
  if (blockIdx.x < 6) {               // block-uniform fast path
#pragma unroll
    for (int k = 0; k < RNK; k += 4) {
      int kg = k + 2 * half;
      v2f a = *(const v2f*)&sA[l16 * 32 + kg];
      v2f bv;
      bv.x = D[(long)kg * EIN + n];
      bv.y = D[(long)(kg + 1) * EIN + n];
      acc = WMMA_F32(a, bv, acc);
    }
  } else {                            // edge block: per-lane guarded loads
#pragma unroll
    for (int k = 0; k < RNK; k += 4) {
      int kg = k + 2 * half;
      v2f a = *(const v2f*)&sA[l16 * 32 + kg];
      v2f bv;
      bv.x = (n < EIN) ? D[(long)kg * EIN + n] : 0.0f;
      bv.y = (n < EIN) ? D[(long)(kg + 1) * EIN + n] : 0.0f;
      acc = WMMA_F32(a, bv, acc);
    }
  }

  // C tile -> LDS (16 x 128 block tile)
#pragma unroll
  for (int v = 0; v < 8; ++v)
    sC[(v + 8 * half) * 128 + wave * 16 + l16] = acc[v];
  __syncthreads();

  // cooperative coalesced stores
  if (blockIdx.x < 6) {
#pragma unroll
    for (int p = 0; p < 4; ++p) {
      int idx = p * 512 + tid * 2;
      int row = idx >> 7, col = idx & 127;
      *(v2f*)&out[(m0 + row) * EIN + blockIdx.x * 128 + col] = *(const v2f*)&sC[idx];
    }
  } else {
#pragma unroll
    for (int i = 0; i < 8; ++i) {
      int idx = tid + i * 256;
      int row = idx >> 7, col = idx & 127;
      int nn = 768 + col;
      if (nn < EIN) out[(m0 + row) * EIN + nn] = sC[idx];
    }
  }
}

// ---------------------------------------------------------------------------
extern "C" void kernel_launch(void* const* d_in, const int* in_sizes, int n_in,
                              void* d_out, int out_size, void* d_ws, size_t ws_size,
                              hipStream_t stream) {
  const float* x_enc  = (const float*)d_in[0];
  const float* pca_en = (const float*)d_in[1];
  const float* pca_de = (const float*)d_in[2];
  const float* Wr     = (const float*)d_in[3];
  const float* Wi     = (const float*)d_in[4];
  const float* br     = (const float*)d_in[5];
  const float* bi     = (const float*)d_in[6];
  float* out = (float*)d_out;

  float* w     = (float*)d_ws;
  float* Y     = w + OFF_Y;      // also Z (aliased after forward DFT)
  float* meanv = w + OFF_MEAN;
  float* stdv  = w + OFF_STD;
  float* istdv = w + OFF_ISTD;
  float* Xr    = w + OFF_XR;
  float* Xi    = w + OFF_XI;
  float* Or    = w + OFF_OR;
  float* Oi    = w + OFF_OI;
  float* Fcos  = w + OFF_FCOS;
  float* Fsin  = w + OFF_FSIN;
  float* Ccat  = w + OFF_CCAT;

  // K0: bases (138240 dominant elements)
  k_basis<<<(TSEQ * 192 + 255) / 256, 256, 0, stream>>>(Fcos, Fsin, Ccat);
  // K1: GEMM1 -> Y  (async-LDS double-buffered)
  k_gemm1<<<MROWS / 64, 256, 0, stream>>>(x_enc, pca_en, Y);
  // K2: stats
  k_stats<<<NB, 256, 0, stream>>>(Y, meanv, stdv, istdv);
  // K3: forward DFT -> Xr, Xi
  k_dft<<<NB, 192, 0, stream>>>(Y, meanv, istdv, Fcos, Fsin, Xr, Xi);
  // K4: complex mixing -> Or, Oi
  k_mix<<<NB * 12, 256, 0, stream>>>(Xr, Xi, Wr, Wi, br, bi, Or, Oi);
  // K5: inverse DFT + de-normalize -> Z (aliases Y)
  k_idft<<<NB, 256, 0, stream>>>(Or, Oi, Ccat, meanv, stdv, Y);
  // K6: GEMM2 -> out (coalesced b64 stores via LDS)
  dim3 g6(7, MROWS / 16);
  k_gemm2<<<g6, 256, 0, stream>>>(Y, pca_de, out);
}